// BoxFilter_2156073583165
// MI455X (gfx1250) — compile-verified
//
#include <hip/hip_runtime.h>
#include <stdint.h>

#define H    1024
#define W    1024
#define TX   64
#define TY   64
#define R    4
#define INW  72          // TX + 2R
#define INH  72          // TY + 2R
#define NTH  256

typedef __attribute__((ext_vector_type(2))) float v2f;
typedef __attribute__((ext_vector_type(4))) float v4f;
typedef __attribute__((ext_vector_type(8))) float v8f;

typedef __attribute__((__vector_size__(16))) int v4i;
typedef __attribute__((address_space(1))) v4i glob_v4i;
typedef __attribute__((address_space(3))) v4i lds_v4i;

#if defined(__AMDGCN__) && __has_builtin(__builtin_amdgcn_global_load_async_to_lds_b128)
#define USE_ASYNC_B128 1
#endif

// Band fragment: value 1 iff m <= k <= m+8, for k = k0 (x) and k0+1 (y).
// Serves as A (16x4, lane m = l&15, K = v + 2*(l>>4)) for the vertical pass and
// as B (4x16, lane n = l&15, K = v + 2*(l>>4)) for the horizontal pass.
__device__ __forceinline__ v2f band_frag(int m, int k0) {
  v2f b;
  b.x = (m <= k0     && k0     <= m + 8) ? 1.0f : 0.0f;
  b.y = (m <= k0 + 1 && k0 + 1 <= m + 8) ? 1.0f : 0.0f;
  return b;
}

__global__ __launch_bounds__(NTH) void box9_kernel(const float* __restrict__ in,
                                                   float* __restrict__ out) {
  __shared__ float s_in[INH * INW];   // halo-staged input tile (72x72)
  __shared__ float s_tmp[TY * INW];   // vertically summed tile (64x72)

  const int tid = threadIdx.x;
  const size_t base = (size_t)blockIdx.z * (size_t)(H * W);
  const int gx0 = blockIdx.x * TX;
  const int gy0 = blockIdx.y * TY;

  const int lane = tid & 31;
  const int wave = tid >> 5;
  const int m    = lane & 15;
  const int half = lane >> 4;

  // ---- Phase 1: stage 72x72 halo tile via async b128 (18 chunks of 4 floats per row)
  // Every 4-aligned chunk (offset -4 from a 64-aligned gx0) is either fully in-row or
  // fully wrapped across the 1024 boundary, so each b128 is contiguous in memory.
  for (int ci = tid; ci < (INH * INW) / 4; ci += NTH) {
    int ry = ci / 18;
    int c4 = (ci - ry * 18) * 4;
    int gy = (gy0 + ry - R) & (H - 1);
    int gx = (gx0 + c4 - R) & (W - 1);
    const float* gp = in + base + (size_t)gy * W + gx;
#ifdef USE_ASYNC_B128
    __builtin_amdgcn_global_load_async_to_lds_b128(
        (glob_v4i*)(uintptr_t)gp,
        (lds_v4i*)(unsigned)(uintptr_t)&s_in[ry * INW + c4],
        0, 0);
#else
    *(v4f*)&s_in[ry * INW + c4] = *(const v4f*)gp;
#endif
  }
#if defined(USE_ASYNC_B128)
#if __has_builtin(__builtin_amdgcn_s_wait_asynccnt)
  __builtin_amdgcn_s_wait_asynccnt(0);
#else
  asm volatile("s_wait_asynccnt 0" ::: "memory");
#endif
#endif
  __syncthreads();

  // ---- Phase 2: vertical 9-tap as banded matmul: D = Band(16x24) * In(24x16).
  // 20 tiles: i0 = (tt&3)*16 (4 row blocks), j0 = (tt>>2)*16 (5 col blocks, last partial).
  for (int tt = wave; tt < 20; tt += 8) {
    const int i0 = (tt & 3) * 16;
    const int j0 = (tt >> 2) * 16;
    v8f c = {};
#pragma unroll
    for (int kk = 0; kk < 6; ++kk) {
      const int k0 = 4 * kk + 2 * half;
      v2f a = band_frag(m, k0);
      // B fragment: lane l, vgpr v holds In[k0+v][l&15] = s_in[(i0+k0+v)*72 + j0 + m]
      v2f b;
      b.x = s_in[(i0 + k0)     * INW + j0 + m];
      b.y = s_in[(i0 + k0 + 1) * INW + j0 + m];
      c = __builtin_amdgcn_wmma_f32_16x16x4_f32(false, a, false, b, (short)0, c,
                                                false, false);
    }
    // D: vgpr r -> row r + 8*half, col m. Mask the partial (j0==64) column tile.
    if (j0 + m < INW) {
#pragma unroll
      for (int r = 0; r < 8; ++r) {
        s_tmp[(i0 + r + 8 * half) * INW + j0 + m] = c[r];
      }
    }
  }
  __syncthreads();

  // ---- Phase 3: horizontal 9-tap as banded matmul: D = Tmp(16x24) * Band(24x16).
  const float inv_area = 1.0f / 81.0f;
  for (int t = 0; t < 2; ++t) {
    const int tile = wave * 2 + t;       // 16 tiles of 16x16 over 64x64
    const int i0 = (tile >> 2) * 16;
    const int j0 = (tile & 3) * 16;

    v8f c = {};
#pragma unroll
    for (int kk = 0; kk < 6; ++kk) {
      const int k0 = 4 * kk + 2 * half;
      // A fragment: lane l, vgpr v holds Tmp[l&15][k0+v] (consecutive -> b64 LDS load)
      const float* tp = &s_tmp[(i0 + m) * INW + j0 + k0];
      v2f a;
      a.x = tp[0];
      a.y = tp[1];
      v2f b = band_frag(m, k0);
      c = __builtin_amdgcn_wmma_f32_16x16x4_f32(false, a, false, b, (short)0, c,
                                                false, false);
    }

    const size_t obase = base + (size_t)(gy0 + i0) * W + (size_t)(gx0 + j0);
#pragma unroll
    for (int r = 0; r < 8; ++r) {
      out[obase + (size_t)(r + 8 * half) * W + m] = c[r] * inv_area;
    }
  }
}

extern "C" void kernel_launch(void* const* d_in, const int* in_sizes, int n_in,
                              void* d_out, int out_size, void* d_ws, size_t ws_size,
                              hipStream_t stream) {
  const float* in = (const float*)d_in[0];
  float* out = (float*)d_out;
  const int nimg = in_sizes[0] / (H * W);   // 8*3 = 24
  dim3 grid(W / TX, H / TY, nimg);
  box9_kernel<<<grid, dim3(NTH, 1, 1), 0, stream>>>(in, out);
}